// ActorModel_86560770884130
// MI455X (gfx1250) — compile-verified
//
#include <hip/hip_runtime.h>

typedef float v2f __attribute__((ext_vector_type(2)));
typedef float v8f __attribute__((ext_vector_type(8)));

#define ROW_STRIDE 34   // 30 spatial cols, padded to 32, +2 to stagger banks (keeps float2 align)

__global__ __launch_bounds__(256) void actor_fused_kernel(
    const float* __restrict__ own,    // [B,3,6]
    const float* __restrict__ ball,   // [B,3,3]
    const float* __restrict__ Wlva,   // [3,3,10,2]
    const float* __restrict__ blva,   // [3,3,10]
    const float* __restrict__ Wnorm,  // [3,10,3]
    const float* __restrict__ Wout,   // [9,30]
    const float* __restrict__ bout,   // [9]
    float* __restrict__ out)          // [B,9]
{
    __shared__ float sw[360];                    // W_lva | b_lva | W_norm
    __shared__ float ssp[8 * 32 * ROW_STRIDE];   // per-wave spatial staging

    const int tid  = threadIdx.x;
    const int wave = tid >> 5;
    const int lane = tid & 31;
    const int n    = lane & 15;   // WMMA column index (output channel)
    const int h    = lane >> 4;   // lane half

    // ---- cooperative broadcast-weight load into LDS ----
    if (tid < 180)        sw[tid] = Wlva[tid];
    else if (tid < 270)   sw[tid] = blva[tid - 180];
    else if (tid < 360)   sw[tid] = Wnorm[tid - 270];

    // ---- per-wave B fragments for the final 30->9 linear (K padded to 32) ----
    v2f bfrag[8];
    #pragma unroll
    for (int c = 0; c < 8; ++c) {
        #pragma unroll
        for (int v = 0; v < 2; ++v) {
            int k = 4 * c + 2 * h + v;
            bfrag[c][v] = (n < 9 && k < 30) ? Wout[n * 30 + k] : 0.0f;
        }
    }
    const float bo = (n < 9) ? bout[n] : 0.0f;

    __syncthreads();

    const long long g = (long long)blockIdx.x * 256 + tid;

    // ---- load this element's inputs (own: 8B-aligned pairs, ball: b32) ----
    float oc[18];
    #pragma unroll
    for (int j = 0; j < 9; ++j) {
        v2f t = *(const v2f*)(own + g * 18 + 2 * j);
        oc[2 * j] = t[0]; oc[2 * j + 1] = t[1];
    }
    float gb[9];
    #pragma unroll
    for (int j = 0; j < 9; ++j) gb[j] = ball[g * 9 + j];

    // ---- VALU: spatial[30] = prod_t(proc) * normal per branch ----
    float sp[30];
    #pragma unroll
    for (int i = 0; i < 3; ++i) {
        #pragma unroll
        for (int o = 0; o < 10; ++o) {
            float prod = 1.0f;
            #pragma unroll
            for (int t = 0; t < 3; ++t) {
                const int it = i * 3 + t;
                float p = sw[(it * 10 + o) * 2 + 0] * oc[i * 6 + t]
                        + sw[(it * 10 + o) * 2 + 1] * gb[i * 3 + t]
                        + sw[180 + it * 10 + o];
                prod *= p;
            }
            float nr = sw[270 + (i * 10 + o) * 3 + 0] * oc[i * 6 + 3]
                     + sw[270 + (i * 10 + o) * 3 + 1] * oc[i * 6 + 4]
                     + sw[270 + (i * 10 + o) * 3 + 2] * oc[i * 6 + 5];
            sp[i * 10 + o] = prod * nr;
        }
    }

    // ---- stage spatial row into this wave's LDS tile (K padded 30->32 with zeros) ----
    float* rowp = ssp + (wave * 32 + lane) * ROW_STRIDE;
    #pragma unroll
    for (int j = 0; j < 15; ++j) {
        v2f t; t[0] = sp[2 * j]; t[1] = sp[2 * j + 1];
        *(v2f*)(rowp + 2 * j) = t;
    }
    { v2f z; z[0] = 0.0f; z[1] = 0.0f; *(v2f*)(rowp + 30) = z; }

    __syncthreads();

    // ---- WMMA: two 16-row tiles per wave, 8 K-chunks each ----
    const float* wbase = ssp + wave * 32 * ROW_STRIDE;
    const long long outBase = (long long)blockIdx.x * 256 + wave * 32;

    #pragma unroll
    for (int T = 0; T < 2; ++T) {
        v8f acc = {};
        #pragma unroll
        for (int c = 0; c < 8; ++c) {
            // A fragment: lane holds (row = T*16+n, K = 4c+2h .. 4c+2h+1)
            v2f a = *(const v2f*)(wbase + (T * 16 + n) * ROW_STRIDE + 4 * c + 2 * h);
            acc = __builtin_amdgcn_wmma_f32_16x16x4_f32(
                false, a, false, bfrag[c], (short)0, acc, false, false);
        }
        // D layout: VGPR v -> row (v + 8h), lane n -> output channel
        if (n < 9) {
            #pragma unroll
            for (int v = 0; v < 8; ++v) {
                long long row = outBase + T * 16 + v + 8 * h;
                out[row * 9 + n] = acc[v] + bo;
            }
        }
    }
}

extern "C" void kernel_launch(void* const* d_in, const int* in_sizes, int n_in,
                              void* d_out, int out_size, void* d_ws, size_t ws_size,
                              hipStream_t stream) {
    const float* own  = (const float*)d_in[0];
    const float* ball = (const float*)d_in[1];
    const float* Wlva = (const float*)d_in[2];
    const float* blva = (const float*)d_in[3];
    const float* Wn   = (const float*)d_in[4];
    const float* Wo   = (const float*)d_in[5];
    const float* bo   = (const float*)d_in[6];
    float* out = (float*)d_out;

    const long long Bn = (long long)in_sizes[0] / 18;  // [B,3,6]
    const int blocks = (int)(Bn / 256);                // B = 2^21, exact
    actor_fused_kernel<<<blocks, 256, 0, stream>>>(own, ball, Wlva, blva, Wn, Wo, bo, out);
}